// StatDenoiser_14791867368133
// MI455X (gfx1250) — compile-verified
//
#include <hip/hip_runtime.h>
#include <hip/hip_bf16.h>

#define H_  256
#define W_  256
#define T_  48
#define HWT (256*256*48)

#define TH 16
#define TW 8
#define TT 8
#define RS 3
#define RT 2
#define HH_ (TH + 2*RS)      // 22
#define HW_ (TW + 2*RS)      // 14
#define HT_ (TT + 2*RT)      // 12  (multiple of 4 -> b128-aligned t-rows)
#define CH_SZ (HH_*HW_*HT_)  // 3696 floats per channel
#define NLOAD 18             // async-filled: 9 guidance + 3 images + 3 est + 3 var
#define NARR  19             // + 1 derived A field (var slots become P in place)
#define SMEM_FLOATS (NARR*CH_SZ)   // 70224 floats = 280,896 B (< 320 KB WGP LDS)

typedef float v2f __attribute__((ext_vector_type(2)));

__device__ __forceinline__ void load8(const float* p, float v[8]) {
    // p is 16-byte aligned by construction (all strides multiples of 4 floats)
    float4 a = ((const float4*)p)[0];
    float4 b = ((const float4*)p)[1];
    v[0] = a.x; v[1] = a.y; v[2] = a.z; v[3] = a.w;
    v[4] = b.x; v[5] = b.y; v[6] = b.z; v[7] = b.w;
}
__device__ __forceinline__ v2f mk2(const float v[8], int k) {
    return (v2f){v[k], v[k + 1]};
}

__global__ void __launch_bounds__(256, 1)
StatDenoiser_kernel(const float* __restrict__ images,
                    const float* __restrict__ guidance,
                    const float* __restrict__ estimands,
                    const float* __restrict__ variance,
                    const float* __restrict__ sigma_inv,
                    const int*   __restrict__ spp,
                    float* __restrict__ out)
{
    extern __shared__ float smem[];

    const int tid = threadIdx.x;
    const int t0  = blockIdx.x * TT;
    const int w0  = blockIdx.y * TW;
    const int h0  = blockIdx.z * TH;

    // ---------------- Phase 1: fill LDS halo tile (async global -> LDS DMA) ----
    for (int i = tid; i < NLOAD * CH_SZ; i += 256) {
        int ch = i / CH_SZ;
        int r  = i - ch * CH_SZ;
        int hh = r / (HW_*HT_);
        int r2 = r - hh * (HW_*HT_);
        int ww = r2 / HT_;
        int lt = r2 - ww * HT_;
        int gh = h0 - RS + hh;
        int gw = w0 - RS + ww;
        int gt = t0 - RT + lt;

        const float* base;
        float padv;
        if      (ch <  9) { base = guidance  +  ch      * HWT; padv =  0.0f; }
        else if (ch < 12) { base = images    + (ch -  9)* HWT; padv =  0.0f; }
        else if (ch < 15) { base = estimands + (ch - 12)* HWT; padv = -2.0f; } // pad = -2
        else              { base = variance  + (ch - 15)* HWT; padv =  0.0f; }

        bool inb = ((unsigned)gh < (unsigned)H_) &
                   ((unsigned)gw < (unsigned)W_) &
                   ((unsigned)gt < (unsigned)T_);
        if (inb) {
            const float* g = base + ((size_t)gh * W_ + gw) * T_ + gt;
            unsigned ldsoff = (unsigned)(size_t)(&smem[i]);   // addr[31:0] == LDS byte offset
            asm volatile("global_load_async_to_lds_b32 %0, %1, off"
                         :: "v"(ldsoff), "v"(g) : "memory");
        } else {
            smem[i] = padv;   // disjoint LDS slots -> no ordering hazard vs async DMA
        }
    }
    asm volatile("s_wait_asynccnt 0" ::: "memory");
    __syncthreads();

    // ---------------- Scalars -------------------------------------------------
    const float LOG2E = 1.4426950408889634f;
    const float T2c   = 7.8794385769f;                       // Z_CRIT^2
    float sig2[9];
    #pragma unroll
    for (int k = 0; k < 9; ++k) sig2[k] = -sigma_inv[k] * LOG2E;  // fold neg + ln2
    const float sppf = (float)spp[0];

    // ---------------- Phase 2: derived per-voxel fields in LDS ----------------
    //   A(x)  = sum_c sig2_c * g_c(x)^2            -> array 18
    //   P_c(x)= spp*e_c(x)^2 - T2*v_c(x)  (in place over variance slots 15..17)
    for (int i = tid; i < CH_SZ; i += 256) {
        float a = 0.0f;
        #pragma unroll
        for (int c = 0; c < 9; ++c) {
            float g = smem[c * CH_SZ + i];
            a = fmaf(sig2[c] * g, g, a);
        }
        smem[18 * CH_SZ + i] = a;
        #pragma unroll
        for (int c = 0; c < 3; ++c) {
            float e = smem[(12 + c) * CH_SZ + i];
            float v = smem[(15 + c) * CH_SZ + i];
            smem[(15 + c) * CH_SZ + i] = fmaf(sppf * e, e, -(T2c * v));
        }
    }
    __syncthreads();

    // ---------------- Per-thread setup ----------------------------------------
    const int tq  = tid & 1;            // which t-quad (0 or 1) of the 8-wide t tile
    const int wwo = (tid >> 1) & 7;     // output w within tile
    const int hho = tid >> 4;           // output h within tile
    const int ltb = 4 * tq;             // base of the 8-float t window in halo coords

    const int crow = ((hho + RS) * HW_ + (wwo + RS)) * HT_ + ltb;   // center row

    // pre-scaled center terms as packed pairs (j = {0,1} and {2,3})
    v2f sg[9][2];                       // sig2_c * g_center
    #pragma unroll
    for (int c = 0; c < 9; ++c) {
        float v8[8]; load8(&smem[c * CH_SZ + crow], v8);
        sg[c][0] = mk2(v8, 2) * sig2[c];
        sg[c][1] = mk2(v8, 4) * sig2[c];
    }
    v2f Cp[2];                          // A at center
    {
        float v8[8]; load8(&smem[18 * CH_SZ + crow], v8);
        Cp[0] = mk2(v8, 2);
        Cp[1] = mk2(v8, 4);
    }
    v2f m2se[3][2], Qp[3][2];           // -2*spp*e_center ; P at center
    #pragma unroll
    for (int c = 0; c < 3; ++c) {
        float v8[8];
        load8(&smem[(12 + c) * CH_SZ + crow], v8);
        m2se[c][0] = mk2(v8, 2) * (-2.0f * sppf);
        m2se[c][1] = mk2(v8, 4) * (-2.0f * sppf);
        load8(&smem[(15 + c) * CH_SZ + crow], v8);
        Qp[c][0] = mk2(v8, 2);
        Qp[c][1] = mk2(v8, 4);
    }

    const v2f vm2 = {-2.0f, -2.0f};
    v2f num2[3][2], den2[2];
    #pragma unroll
    for (int c = 0; c < 3; ++c) { num2[c][0] = (v2f)0.0f; num2[c][1] = (v2f)0.0f; }
    den2[0] = (v2f)0.0f; den2[1] = (v2f)0.0f;

    // ---------------- 7x7 spatial taps, 5 temporal taps via register window ----
    #pragma unroll 1
    for (int oy = 0; oy < 7; ++oy) {
        #pragma unroll 1
        for (int ox = 0; ox < 7; ++ox) {
            const int nrow = ((hho + oy) * HW_ + (wwo + ox)) * HT_ + ltb;

            // ---- guidance cross terms: X[dt][p] = sum_c sig2*g_cen*g_nbr -----
            v2f X[5][2];
            #pragma unroll
            for (int c = 0; c < 9; ++c) {
                float v8[8]; load8(&smem[c * CH_SZ + nrow], v8);
                #pragma unroll
                for (int dt = 0; dt < 5; ++dt) {
                    if (c == 0) {   // fused init: mul instead of zero+fma
                        X[dt][0] = mk2(v8, dt)     * sg[0][0];
                        X[dt][1] = mk2(v8, dt + 2) * sg[0][1];
                    } else {
                        X[dt][0] = __builtin_elementwise_fma(mk2(v8, dt),     sg[c][0], X[dt][0]);
                        X[dt][1] = __builtin_elementwise_fma(mk2(v8, dt + 2), sg[c][1], X[dt][1]);
                    }
                }
            }

            // ---- nd = A_nbr + A_cen - 2X (already negative & log2-scaled) ----
            // ---- weight = exp2(nd)                                        ----
            v2f wv[5][2];
            {
                float a8[8]; load8(&smem[18 * CH_SZ + nrow], a8);
                #pragma unroll
                for (int dt = 0; dt < 5; ++dt) {
                    #pragma unroll
                    for (int p = 0; p < 2; ++p) {
                        v2f AC = mk2(a8, dt + 2*p) + Cp[p];
                        v2f nd = __builtin_elementwise_fma(vm2, X[dt][p], AC);
                        wv[dt][p].x = __builtin_amdgcn_exp2f(nd.x);   // v_exp_f32
                        wv[dt][p].y = __builtin_amdgcn_exp2f(nd.y);
                    }
                }
            }

            // ---- z-test: s_c = P_nbr + P_cen - 2*spp*en*e_cen; gate on
            // ---- max3(s0,s1,s2) <= 0  (single v_max3_f32 per element) --------
            {
                float e8[3][8], p8[3][8];
                #pragma unroll
                for (int c = 0; c < 3; ++c) {
                    load8(&smem[(12 + c) * CH_SZ + nrow], e8[c]);
                    load8(&smem[(15 + c) * CH_SZ + nrow], p8[c]);
                }
                #pragma unroll
                for (int dt = 0; dt < 5; ++dt) {
                    #pragma unroll
                    for (int p = 0; p < 2; ++p) {
                        v2f s0 = __builtin_elementwise_fma(m2se[0][p], mk2(e8[0], dt + 2*p),
                                                           mk2(p8[0], dt + 2*p) + Qp[0][p]);
                        v2f s1 = __builtin_elementwise_fma(m2se[1][p], mk2(e8[1], dt + 2*p),
                                                           mk2(p8[1], dt + 2*p) + Qp[1][p]);
                        v2f s2 = __builtin_elementwise_fma(m2se[2][p], mk2(e8[2], dt + 2*p),
                                                           mk2(p8[2], dt + 2*p) + Qp[2][p]);
                        float mx = fmaxf(fmaxf(s0.x, s1.x), s2.x);    // v_max3_f32
                        float my = fmaxf(fmaxf(s0.y, s1.y), s2.y);
                        wv[dt][p].x = (mx <= 0.0f) ? wv[dt][p].x : 0.0f;
                        wv[dt][p].y = (my <= 0.0f) ? wv[dt][p].y : 0.0f;
                    }
                }
            }

            // ---- weighted accumulation (packed) ----
            #pragma unroll
            for (int c = 0; c < 3; ++c) {
                float i8[8]; load8(&smem[(9 + c) * CH_SZ + nrow], i8);
                #pragma unroll
                for (int dt = 0; dt < 5; ++dt) {
                    num2[c][0] = __builtin_elementwise_fma(wv[dt][0], mk2(i8, dt),     num2[c][0]);
                    num2[c][1] = __builtin_elementwise_fma(wv[dt][1], mk2(i8, dt + 2), num2[c][1]);
                }
            }
            #pragma unroll
            for (int dt = 0; dt < 5; ++dt) {
                den2[0] += wv[dt][0];
                den2[1] += wv[dt][1];
            }
        }
    }

    // ---------------- Store (4 consecutive t -> one b128 store) ----------------
    const int gh = h0 + hho, gw = w0 + wwo, gt = t0 + ltb;
    #pragma unroll
    for (int c = 0; c < 3; ++c) {
        float4 o;
        o.x = num2[c][0].x / den2[0].x;
        o.y = num2[c][0].y / den2[0].y;
        o.z = num2[c][1].x / den2[1].x;
        o.w = num2[c][1].y / den2[1].y;
        *(float4*)(out + ((size_t)(c * H_ + gh) * W_ + gw) * T_ + gt) = o;
    }
}

extern "C" void kernel_launch(void* const* d_in, const int* in_sizes, int n_in,
                              void* d_out, int out_size, void* d_ws, size_t ws_size,
                              hipStream_t stream) {
    const float* images    = (const float*)d_in[0];
    const float* guidance  = (const float*)d_in[1];
    const float* estimands = (const float*)d_in[2];
    const float* variance  = (const float*)d_in[3];
    const float* sigma_inv = (const float*)d_in[4];
    const int*   spp       = (const int*)d_in[5];
    float*       out       = (float*)d_out;

    const size_t smem = (size_t)SMEM_FLOATS * sizeof(float);   // 280,896 B
    (void)hipFuncSetAttribute((const void*)StatDenoiser_kernel,
                              hipFuncAttributeMaxDynamicSharedMemorySize, (int)smem);

    dim3 grid(T_ / TT, W_ / TW, H_ / TH);   // (6, 32, 16)
    dim3 block(256);
    StatDenoiser_kernel<<<grid, block, smem, stream>>>(
        images, guidance, estimands, variance, sigma_inv, spp, out);
}